// NL_model_86835648791026
// MI455X (gfx1250) — compile-verified
//
#include <hip/hip_runtime.h>
#include <math.h>

typedef __attribute__((ext_vector_type(2))) float v2f;
typedef __attribute__((ext_vector_type(8))) float v8f;

#define N_NODES 100000
#define N_EDGES 1600000
#define ATTR 16
#define HID 32
#define NBASIS 8
#define SH_DIM 4
#define R_CUT 5.0f
#define PI_F 3.14159265358979323846f

// Sum over lane-groups-of-4 via DPP8 (pure VALU, no LDS traffic).
// sel xor1 = lanes (1,0,3,2,5,4,7,6), sel xor2 = lanes (2,3,0,1,6,7,4,5).
__device__ __forceinline__ float group4_sum(float v) {
  float v1 = __int_as_float(
      __builtin_amdgcn_mov_dpp8(__float_as_int(v), 14570689));  // xor 1
  float s = v + v1;
  float v2 = __int_as_float(
      __builtin_amdgcn_mov_dpp8(__float_as_int(s), 11788826));  // xor 2
  return s + v2;
}

// ---------------------------------------------------------------------------
// Kernel 1: node embedding  h0 = elem_embed[x] @ W_embed ; also stash x_attr
// ---------------------------------------------------------------------------
__global__ void __launch_bounds__(256) embed_kernel(
    const int* __restrict__ x, const float* __restrict__ elem_embed,
    const float* __restrict__ W_embed, float* __restrict__ x_attr,
    float* __restrict__ h) {
  int node = blockIdx.x * blockDim.x + threadIdx.x;
  if (node >= N_NODES) return;
  int sp = x[node];
  float a[ATTR];
#pragma unroll
  for (int i = 0; i < ATTR; ++i) {
    a[i] = elem_embed[sp * ATTR + i];
    x_attr[node * ATTR + i] = a[i];
  }
#pragma unroll 4
  for (int j = 0; j < HID; ++j) {
    float acc = 0.f;
#pragma unroll
    for (int i = 0; i < ATTR; ++i) acc = fmaf(a[i], W_embed[i * HID + j], acc);
    h[node * HID + j] = acc;
  }
}

// ---------------------------------------------------------------------------
// Kernel 2: zero the aggregation buffer
// ---------------------------------------------------------------------------
__global__ void __launch_bounds__(256) zero_kernel(float* __restrict__ p, int n) {
  int i = blockIdx.x * blockDim.x + threadIdx.x;
  if (i < n) p[i] = 0.f;
}

// ---------------------------------------------------------------------------
// Kernel 3: edge messages via WMMA f32 16x16x4.
// Block = 256 threads (8 waves); block handles 128 edges.
// Stage 1: threads 0..127 compute geometry -> LDS (sh[4], rbf[8], src, dst).
// Stage 2: each wave owns 16 edges; computes w = rbf@Wr + br with WMMA
//          (M=16 edges, N=128 cols in 8 tiles, K=8 as two K=4 steps),
//          contracts with edge_sh via DPP8 lane-group-of-4 reduction,
//          multiplies by gathered h[src], atomically scatters to agg[dst].
// ---------------------------------------------------------------------------
__global__ void __launch_bounds__(256) edge_msg_kernel(
    const float* __restrict__ pos, const int* __restrict__ edge_index,
    const int* __restrict__ cell_offset, const float* __restrict__ cell,
    const float* __restrict__ h_in, const float* __restrict__ Wr_l,
    const float* __restrict__ br_l, float* __restrict__ agg) {
  __shared__ float s_sh[128][SH_DIM];
  __shared__ float s_rbf[128][NBASIS];
  __shared__ int s_src[128];
  __shared__ int s_dst[128];

  const int tid = threadIdx.x;
  const int base = blockIdx.x * 128;

  // ---- Stage 1: geometry + radial basis into LDS ----
  if (tid < 128) {
    int e = base + tid;
    if (e < N_EDGES) {
      int srcn = edge_index[e];
      int dstn = edge_index[N_EDGES + e];
      s_src[tid] = srcn;
      s_dst[tid] = dstn;
      float ox = (float)cell_offset[e * 3 + 0] - 1.0f;
      float oy = (float)cell_offset[e * 3 + 1] - 1.0f;
      float oz = (float)cell_offset[e * 3 + 2] - 1.0f;
      float pvx = ox * cell[0] + oy * cell[3] + oz * cell[6];
      float pvy = ox * cell[1] + oy * cell[4] + oz * cell[7];
      float pvz = ox * cell[2] + oy * cell[5] + oz * cell[8];
      float ex = pos[dstn * 3 + 0] - pos[srcn * 3 + 0] + pvx;
      float ey = pos[dstn * 3 + 1] - pos[srcn * 3 + 1] + pvy;
      float ez = pos[dstn * 3 + 2] - pos[srcn * 3 + 2] + pvz;
      float r = sqrtf(ex * ex + ey * ey + ez * ez);
      float r_safe = fmaxf(r, 1e-9f);
      float inv = 1.0f / r_safe;
      s_sh[tid][0] = 1.0f;
      s_sh[tid][1] = ex * inv;
      s_sh[tid][2] = ey * inv;
      s_sh[tid][3] = ez * inv;
      // polynomial envelope, p = 6
      float u = r * (1.0f / R_CUT);
      float u2 = u * u;
      float u4 = u2 * u2;
      float u6 = u4 * u2;
      float u7 = u6 * u;
      float u8 = u4 * u4;
      float env = 1.0f - 28.0f * u6 + 48.0f * u7 - 21.0f * u8;
      env = (u < 1.0f) ? env : 0.0f;
      float k = env * inv;
#pragma unroll
      for (int n = 0; n < NBASIS; ++n)
        s_rbf[tid][n] = __sinf((float)(n + 1) * (PI_F / R_CUT) * r) * k;
    } else {
      s_src[tid] = 0;
      s_dst[tid] = 0;
#pragma unroll
      for (int s = 0; s < SH_DIM; ++s) s_sh[tid][s] = 0.f;
#pragma unroll
      for (int n = 0; n < NBASIS; ++n) s_rbf[tid][n] = 0.f;
    }
  }
  __syncthreads();

  // ---- Stage 2: WMMA per wave ----
  const int lane = tid & 31;
  const int w = tid >> 5;        // wave id in block (wave32)
  const int m_lo = lane & 15;    // A/B: M (edge) or N column sub-index
  const int kh = lane >> 4;      // 0: K pair {0,1}; 1: K pair {2,3}
  const int sidx = lane & 3;     // this lane's edge_sh component
  const int jlo = 2 * sidx;      // this lane commits D-VGPRs jlo, jlo+1

  const int le0 = w * 16 + m_lo; // local edge for the A fragment
  v2f a0, a1;
  a0.x = s_rbf[le0][2 * kh];
  a0.y = s_rbf[le0][2 * kh + 1];
  a1.x = s_rbf[le0][4 + 2 * kh];
  a1.y = s_rbf[le0][4 + 2 * kh + 1];

  // Per-lane edge_sh values for the 8 D-VGPRs' edges (hoisted out of t-loop).
  float shv[8];
#pragma unroll
  for (int j = 0; j < 8; ++j) {
    int m = (lane < 16) ? j : (8 + j);
    shv[j] = s_sh[w * 16 + m][sidx];
  }

#pragma unroll
  for (int t = 0; t < 8; ++t) {
    const int col = t * 16 + m_lo;   // column in [0,128)
    v2f b0, b1;
    b0.x = Wr_l[(2 * kh) * 128 + col];
    b0.y = Wr_l[(2 * kh + 1) * 128 + col];
    b1.x = Wr_l[(4 + 2 * kh) * 128 + col];
    b1.y = Wr_l[(5 + 2 * kh) * 128 + col];
    const float brv = br_l[col];
    v8f c;
#pragma unroll
    for (int j = 0; j < 8; ++j) c[j] = brv;  // bias via C operand
    c = __builtin_amdgcn_wmma_f32_16x16x4_f32(false, a0, false, b0,
                                              (short)0, c, false, false);
    c = __builtin_amdgcn_wmma_f32_16x16x4_f32(false, a1, false, b1,
                                              (short)0, c, false, false);
    // contract with edge_sh over s: multiply, then DPP8 group-of-4 reduction
    v8f p;
#pragma unroll
    for (int j = 0; j < 8; ++j) p[j] = group4_sum(shv[j] * c[j]);

    const int hcol = t * 4 + (m_lo >> 2);    // output channel in [0,32)
#pragma unroll
    for (int jj = 0; jj < 2; ++jj) {
      int j = jlo + jj;
      int m = (lane < 16) ? j : (8 + j);
      int lej = w * 16 + m;
      int eg = base + lej;
      if (eg < N_EDGES) {
        int srcn = s_src[lej];
        int dstn = s_dst[lej];
        float msg = h_in[srcn * HID + hcol] * p[j];
        atomicAdd(&agg[dstn * HID + hcol], msg);
      }
    }
  }
}

// ---------------------------------------------------------------------------
// Kernel 4: node update  h = silu(agg @ Wout + x_attr @ Wsc)  via WMMA.
// Wave handles 16 nodes; N=32 in 2 tiles; K=32 (8 steps) + K=16 (4 steps).
// ---------------------------------------------------------------------------
__global__ void __launch_bounds__(256) node_update_kernel(
    const float* __restrict__ agg, const float* __restrict__ x_attr,
    const float* __restrict__ Wout_l, const float* __restrict__ Wsc_l,
    float* __restrict__ h) {
  const int tid = threadIdx.x;
  const int lane = tid & 31;
  const int w = tid >> 5;
  const int nbase = blockIdx.x * 128 + w * 16;
  if (nbase + 16 > N_NODES) return;  // wave-uniform guard (N_NODES % 16 == 0)

  const int m_lo = lane & 15;
  const int kh = lane >> 4;
  const int node = nbase + m_lo;

#pragma unroll
  for (int t = 0; t < 2; ++t) {
    const int col = t * 16 + m_lo;
    v8f c = {0.f, 0.f, 0.f, 0.f, 0.f, 0.f, 0.f, 0.f};
#pragma unroll
    for (int kk = 0; kk < 8; ++kk) {   // agg @ Wout, K = 32
      int k = kk * 4 + 2 * kh;
      v2f a, b;
      a.x = agg[node * HID + k];
      a.y = agg[node * HID + k + 1];
      b.x = Wout_l[k * HID + col];
      b.y = Wout_l[(k + 1) * HID + col];
      c = __builtin_amdgcn_wmma_f32_16x16x4_f32(false, a, false, b,
                                                (short)0, c, false, false);
    }
#pragma unroll
    for (int kk = 0; kk < 4; ++kk) {   // x_attr @ Wsc, K = 16
      int k = kk * 4 + 2 * kh;
      v2f a, b;
      a.x = x_attr[node * ATTR + k];
      a.y = x_attr[node * ATTR + k + 1];
      b.x = Wsc_l[k * HID + col];
      b.y = Wsc_l[(k + 1) * HID + col];
      c = __builtin_amdgcn_wmma_f32_16x16x4_f32(false, a, false, b,
                                                (short)0, c, false, false);
    }
#pragma unroll
    for (int j = 0; j < 8; ++j) {
      int m = (lane < 16) ? j : (8 + j);
      float v = c[j];
      float sv = v / (1.0f + __expf(-v));  // silu
      h[(nbase + m) * HID + col] = sv;
    }
  }
}

// ---------------------------------------------------------------------------
// Kernel 5: readout  out4 = silu(h @ P1) @ P2 ; split energies / forces
// ---------------------------------------------------------------------------
__global__ void __launch_bounds__(256) readout_kernel(
    const float* __restrict__ h, const float* __restrict__ P1,
    const float* __restrict__ P2, float* __restrict__ out) {
  __shared__ float sP1[HID * 32];
  __shared__ float sP2[32 * 4];
  for (int i = threadIdx.x; i < HID * 32; i += blockDim.x) sP1[i] = P1[i];
  for (int i = threadIdx.x; i < 32 * 4; i += blockDim.x) sP2[i] = P2[i];
  __syncthreads();

  int node = blockIdx.x * blockDim.x + threadIdx.x;
  if (node >= N_NODES) return;
  float hr[HID];
#pragma unroll
  for (int k = 0; k < HID; ++k) hr[k] = h[node * HID + k];
  float o0 = 0.f, o1 = 0.f, o2 = 0.f, o3 = 0.f;
#pragma unroll 4
  for (int j = 0; j < 32; ++j) {
    float acc = 0.f;
#pragma unroll
    for (int k = 0; k < HID; ++k) acc = fmaf(hr[k], sP1[k * 32 + j], acc);
    float tv = acc / (1.0f + __expf(-acc));
    o0 = fmaf(tv, sP2[j * 4 + 0], o0);
    o1 = fmaf(tv, sP2[j * 4 + 1], o1);
    o2 = fmaf(tv, sP2[j * 4 + 2], o2);
    o3 = fmaf(tv, sP2[j * 4 + 3], o3);
  }
  out[node] = o0;                          // energies
  out[N_NODES + node * 3 + 0] = o1;        // forces
  out[N_NODES + node * 3 + 1] = o2;
  out[N_NODES + node * 3 + 2] = o3;
}

// ---------------------------------------------------------------------------
extern "C" void kernel_launch(void* const* d_in, const int* in_sizes, int n_in,
                              void* d_out, int out_size, void* d_ws,
                              size_t ws_size, hipStream_t stream) {
  const int* x = (const int*)d_in[0];
  const float* pos = (const float*)d_in[1];
  const int* edge_index = (const int*)d_in[2];
  const float* cell = (const float*)d_in[3];
  const int* cell_offset = (const int*)d_in[4];
  const float* elem_embed = (const float*)d_in[5];
  const float* W_embed = (const float*)d_in[6];
  const float* Wr = (const float*)d_in[7];    // [3][8][128]
  const float* br = (const float*)d_in[8];    // [3][128]
  const float* Wout = (const float*)d_in[9];  // [3][32][32]
  const float* Wsc = (const float*)d_in[10];  // [3][16][32]
  const float* P1 = (const float*)d_in[11];   // [32][32]
  const float* P2 = (const float*)d_in[12];   // [32][4]
  float* out = (float*)d_out;

  // Workspace layout (floats): x_attr | h | agg   = 32 MB total
  float* x_attr = (float*)d_ws;
  float* h = x_attr + (size_t)N_NODES * ATTR;
  float* agg = h + (size_t)N_NODES * HID;

  embed_kernel<<<(N_NODES + 255) / 256, 256, 0, stream>>>(x, elem_embed,
                                                          W_embed, x_attr, h);

  for (int l = 0; l < 3; ++l) {
    zero_kernel<<<(N_NODES * HID + 255) / 256, 256, 0, stream>>>(agg,
                                                                 N_NODES * HID);
    edge_msg_kernel<<<N_EDGES / 128, 256, 0, stream>>>(
        pos, edge_index, cell_offset, cell, h, Wr + l * NBASIS * HID * SH_DIM,
        br + l * HID * SH_DIM, agg);
    node_update_kernel<<<(N_NODES + 127) / 128, 256, 0, stream>>>(
        agg, x_attr, Wout + l * HID * HID, Wsc + l * ATTR * HID, h);
  }

  readout_kernel<<<(N_NODES + 255) / 256, 256, 0, stream>>>(h, P1, P2, out);
}